// GCN_76141180224087
// MI455X (gfx1250) — compile-verified
//
#include <hip/hip_runtime.h>
#include <math.h>

// ---------------------------------------------------------------------------
// GCN forward (3-layer GraphConv + softmax), fp32, CDNA5 WMMA f32 16x16x4.
// ---------------------------------------------------------------------------

typedef __attribute__((ext_vector_type(2))) float v2f;
typedef __attribute__((ext_vector_type(8))) float v8f;

// Explicit global-address-space pointers so loads/stores select
// global_load_* / global_store_* (not flat_*, which also burns DScnt).
#define AS1 __attribute__((address_space(1)))
typedef const AS1 float* gcfp;
typedef       AS1 float* gfp;
typedef const AS1 v2f*   gcv2p;

#define F_IN 512
#define H1D  500
#define H2D  100
#define CD   16

// -------------------- degree / norm --------------------------------------

__global__ void deg_init_kernel(float* __restrict__ deg, int N) {
    int i = blockIdx.x * blockDim.x + threadIdx.x;
    if (i < N) deg[i] = 1.0f;               // self-loop contribution
}

__global__ void deg_accum_kernel(const int* __restrict__ rows,
                                 float* __restrict__ deg, int E) {
    int e = blockIdx.x * blockDim.x + threadIdx.x;
    if (e < E) atomicAdd(&deg[rows[e]], 1.0f);
}

__global__ void dinv_kernel(float* __restrict__ deg, int N) {
    int i = blockIdx.x * blockDim.x + threadIdx.x;
    if (i < N) {
        float d = deg[i];                   // >= 1 (self loop), matches ref mask
        deg[i] = (d > 0.0f) ? rsqrtf(d) : 0.0f;
    }
}

// -------------------- WMMA GEMM: H = A[M,K] * W[K,NCOL] + bias -----------
// One wave computes a 32x32 tile: MT=2 row tiles x NTC=2 col tiles, 4 WMMA
// accumulators.  Fragment layouts (ISA 7.12.2, f32 16x16x4):
//   A: lane = half*16 + l16 loads A[m = l16][k + 2*half .. +1]  (2 VGPRs)
//   B: VGPR0 = rows {k, k+2}, VGPR1 = rows {k+1, k+3}, N = l16
//   C/D: VGPR i -> row i + 8*half, N = l16
// Out-of-range rows/cols are CLAMPED for loads (their outputs are never
// stored, and element (m,n) depends only on A row m / B col n), which keeps
// the K-loop free of exec-mask branches.  K % 4 == 0 for all layers.

#define MT 2
#define NTC 2

template <int K, int NCOL>
__global__ void __launch_bounds__(32)
gemm_bias_wmma(const float* __restrict__ A_, const float* __restrict__ W_,
               const float* __restrict__ bias_, float* __restrict__ H_,
               int M)
{
    gcfp A    = (gcfp)A_;
    gcfp W    = (gcfp)W_;
    gcfp bias = (gcfp)bias_;
    gfp  H    = (gfp)H_;

    const int lane = threadIdx.x & 31;
    const int half = lane >> 4;            // 0 or 1
    const int l16  = lane & 15;

    const int rowBase = blockIdx.x * (16 * MT);
    const int colBase = blockIdx.y * (16 * NTC);

    // Clamped per-lane A-row offsets (include K-phase 2*half).
    size_t aoff[MT];
#pragma unroll
    for (int s = 0; s < MT; ++s) {
        int m = rowBase + s * 16 + l16;
        if (m >= M) m = M - 1;
        aoff[s] = (size_t)m * K + 2 * half;
    }
    // Clamped per-lane B-column offsets (include K-phase 2*half).
    size_t boff[NTC];
#pragma unroll
    for (int t = 0; t < NTC; ++t) {
        int n = colBase + t * 16 + l16;
        if (n >= NCOL) n = NCOL - 1;
        boff[t] = (size_t)(2 * half) * NCOL + n;
    }

    v8f acc[MT][NTC];
#pragma unroll
    for (int s = 0; s < MT; ++s)
#pragma unroll
        for (int t = 0; t < NTC; ++t)
            acc[s][t] = (v8f){0.f,0.f,0.f,0.f,0.f,0.f,0.f,0.f};

    for (int k = 0; k < K; k += 4) {
        v2f a[MT];
#pragma unroll
        for (int s = 0; s < MT; ++s)
            a[s] = *(gcv2p)(A + aoff[s]);  // 8B-aligned: m*K + 2*half is even

        v2f b[NTC];
#pragma unroll
        for (int t = 0; t < NTC; ++t) {
            b[t].x = W[boff[t]];
            b[t].y = W[boff[t] + NCOL];
        }

#pragma unroll
        for (int s = 0; s < MT; ++s)
#pragma unroll
            for (int t = 0; t < NTC; ++t)
                acc[s][t] = __builtin_amdgcn_wmma_f32_16x16x4_f32(
                    false, a[s], false, b[t], (short)0, acc[s][t], false, false);

#pragma unroll
        for (int s = 0; s < MT; ++s) aoff[s] += 4;
#pragma unroll
        for (int t = 0; t < NTC; ++t) boff[t] += 4 * (size_t)NCOL;
    }

#pragma unroll
    for (int t = 0; t < NTC; ++t) {
        const int n = colBase + t * 16 + l16;
        if (n >= NCOL) continue;
        const float bv = bias[n];
#pragma unroll
        for (int s = 0; s < MT; ++s) {
            const int rb = rowBase + s * 16 + 8 * half;
#pragma unroll
            for (int i = 0; i < 8; ++i) {
                const int r = rb + i;
                if (r < M) H[(size_t)r * NCOL + n] = acc[s][t][i] + bv;
            }
        }
    }
}

// -------------------- aggregation ----------------------------------------

// out[i][f] = dinv[i]^2 * h[i][f]   (self-loop term; also initializes out)
__global__ void selfloop_scale_kernel(const float* __restrict__ h_,
                                      const float* __restrict__ dinv_,
                                      float* __restrict__ out_,
                                      int N, int F) {
    gcfp h    = (gcfp)h_;
    gcfp dinv = (gcfp)dinv_;
    gfp  out  = (gfp)out_;
    int idx = blockIdx.x * blockDim.x + threadIdx.x;
    if (idx >= N * F) return;
    int i = idx / F;
    float di = dinv[i];
    out[idx] = di * di * h[idx];
}

// out[row[e]][:] += dinv[row]*dinv[col] * h[col[e]][:]   (one wave per edge)
template <int F>
__global__ void __launch_bounds__(256)
spmm_edges_kernel(const int* __restrict__ rows, const int* __restrict__ cols,
                  const float* __restrict__ dinv_, const float* __restrict__ h_,
                  float* __restrict__ out, int E)
{
    gcfp dinv = (gcfp)dinv_;
    gcfp h    = (gcfp)h_;
    int gid  = blockIdx.x * blockDim.x + threadIdx.x;
    int e    = gid >> 5;
    int lane = gid & 31;
    if (e >= E) return;
    int r = rows[e];
    int c = cols[e];
    float nrm = dinv[r] * dinv[c];
    gcfp hc      = h + (size_t)c * F;
    float* orow  = out + (size_t)r * F;
#pragma unroll
    for (int fb = 0; fb < F; fb += 32) {
        int f = fb + lane;
        if (f < F) atomicAdd(&orow[f], nrm * hc[f]);
    }
}

__global__ void tanh_kernel(float* __restrict__ a_, int n) {
    gfp a = (gfp)a_;
    int i = blockIdx.x * blockDim.x + threadIdx.x;
    if (i < n) a[i] = tanhf(a[i]);
}

// -------------------- final tanh + softmax over C=16 ---------------------

__global__ void tanh_softmax16_kernel(const float* __restrict__ in_,
                                      float* __restrict__ out_, int N) {
    gcfp in  = (gcfp)in_;
    gfp  out = (gfp)out_;
    int i = blockIdx.x * blockDim.x + threadIdx.x;
    if (i >= N) return;
    float v[CD];
    float mx = -3.402823466e38f;
#pragma unroll
    for (int j = 0; j < CD; ++j) {
        v[j] = tanhf(in[i * CD + j]);
        mx = fmaxf(mx, v[j]);
    }
    float s = 0.0f;
#pragma unroll
    for (int j = 0; j < CD; ++j) { v[j] = expf(v[j] - mx); s += v[j]; }
    float inv = 1.0f / s;
#pragma unroll
    for (int j = 0; j < CD; ++j) out[i * CD + j] = v[j] * inv;
}

// -------------------- host-side orchestration ----------------------------

static inline int ceil_div(long long a, long long b) { return (int)((a + b - 1) / b); }

extern "C" void kernel_launch(void* const* d_in, const int* in_sizes, int n_in,
                              void* d_out, int out_size, void* d_ws, size_t ws_size,
                              hipStream_t stream) {
    const float* x   = (const float*)d_in[0];
    const int*   ei  = (const int*)d_in[1];   // [2, E] int32
    const float* W1  = (const float*)d_in[2];
    const float* b1  = (const float*)d_in[3];
    const float* W2  = (const float*)d_in[4];
    const float* b2  = (const float*)d_in[5];
    const float* W3  = (const float*)d_in[6];
    const float* b3  = (const float*)d_in[7];
    float*       out = (float*)d_out;

    const int N = in_sizes[0] / F_IN;
    const int E = in_sizes[1] / 2;
    const int* rows = ei;
    const int* cols = ei + E;

    // Workspace layout (bytes):
    //  [0, 204800)                       : deg / dinv (N floats)
    //  [204800, 204800+100MB)            : bufH  (H1 [N,500]; later H2,H3)
    //  [204800+100MB, 204800+200MB)      : bufA  (agg1/X2 [N,500]);
    //       bufB = bufA (agg2/X3 [N,100]), bufC = bufA+20MB (agg3 [N,16])
    char*  ws   = (char*)d_ws;
    float* deg  = (float*)ws;
    float* bufH = (float*)(ws + 204800);
    float* bufA = (float*)(ws + 204800 + 100000000ull);
    float* bufB = bufA;
    float* bufC = (float*)((char*)bufA + 20000000ull);

    const int TB = 256;

    // ---- symmetric degree norm -------------------------------------------
    deg_init_kernel<<<ceil_div(N, TB), TB, 0, stream>>>(deg, N);
    deg_accum_kernel<<<ceil_div(E, TB), TB, 0, stream>>>(rows, deg, E);
    dinv_kernel<<<ceil_div(N, TB), TB, 0, stream>>>(deg, N);
    const float* dinv = deg;

    const int rowBlocks = ceil_div(N, 16 * MT);

    // ---- layer 1: X[N,512] @ W1[512,500] + b1 -> bufH --------------------
    {
        dim3 grid(rowBlocks, ceil_div(H1D, 16 * NTC));
        gemm_bias_wmma<F_IN, H1D><<<grid, 32, 0, stream>>>(x, W1, b1, bufH, N);
        selfloop_scale_kernel<<<ceil_div((long long)N * H1D, TB), TB, 0, stream>>>(
            bufH, dinv, bufA, N, H1D);
        spmm_edges_kernel<H1D><<<ceil_div((long long)E * 32, TB), TB, 0, stream>>>(
            rows, cols, dinv, bufH, bufA, E);
        tanh_kernel<<<ceil_div((long long)N * H1D, TB), TB, 0, stream>>>(
            bufA, N * H1D);
    }

    // ---- layer 2: bufA[N,500] @ W2[500,100] + b2 -> bufH -----------------
    {
        dim3 grid(rowBlocks, ceil_div(H2D, 16 * NTC));
        gemm_bias_wmma<H1D, H2D><<<grid, 32, 0, stream>>>(bufA, W2, b2, bufH, N);
        selfloop_scale_kernel<<<ceil_div((long long)N * H2D, TB), TB, 0, stream>>>(
            bufH, dinv, bufB, N, H2D);
        spmm_edges_kernel<H2D><<<ceil_div((long long)E * 32, TB), TB, 0, stream>>>(
            rows, cols, dinv, bufH, bufB, E);
        tanh_kernel<<<ceil_div((long long)N * H2D, TB), TB, 0, stream>>>(
            bufB, N * H2D);
    }

    // ---- layer 3: bufB[N,100] @ W3[100,16] + b3 -> bufH ------------------
    {
        dim3 grid(rowBlocks, ceil_div(CD, 16 * NTC));
        gemm_bias_wmma<H2D, CD><<<grid, 32, 0, stream>>>(bufB, W3, b3, bufH, N);
        selfloop_scale_kernel<<<ceil_div((long long)N * CD, TB), TB, 0, stream>>>(
            bufH, dinv, bufC, N, CD);
        spmm_edges_kernel<CD><<<ceil_div((long long)E * 32, TB), TB, 0, stream>>>(
            rows, cols, dinv, bufH, bufC, E);
    }

    // ---- tanh + softmax over C=16 into d_out -----------------------------
    tanh_softmax16_kernel<<<ceil_div(N, TB), TB, 0, stream>>>(bufC, out, N);
}